// ConvLayer_19241453486798
// MI455X (gfx1250) — compile-verified
//
#include <hip/hip_runtime.h>

typedef __attribute__((ext_vector_type(2))) float v2f;
typedef __attribute__((ext_vector_type(8))) float v8f;

#define MD 128      // feature dim M
#define EPB 32      // rows (edges/nodes) per block
#define KSTEP 32    // K-panel depth
#define LDA 36      // padded LDS stride for A panel (float4-aligned, conflict-free)
#define LDH 132     // padded LDS stride for inter-layer H panel

__device__ __forceinline__ v8f wmma4(v2f a, v2f b, v8f c) {
  // D(16x16,f32) = A(16x4,f32) * B(4x16,f32) + C
  return __builtin_amdgcn_wmma_f32_16x16x4_f32(false, a, false, b, (short)0, c,
                                               false, false);
}

__device__ __forceinline__ float lrelu(float x) { return x >= 0.0f ? x : 0.2f * x; }

__device__ __forceinline__ v8f vzero() {
  v8f z = {0.f, 0.f, 0.f, 0.f, 0.f, 0.f, 0.f, 0.f};
  return z;
}

// low 32 bits of a generic pointer to __shared__ = LDS byte offset
__device__ __forceinline__ unsigned lds_off32(const void* p) {
  return (unsigned)(unsigned long long)(uintptr_t)p;
}

// CDNA5 async global->LDS copy, 16B per lane, tracked by ASYNCcnt (ISA 08 §4)
__device__ __forceinline__ void async_b128(unsigned lds_off, const void* gaddr) {
  asm volatile("global_load_async_to_lds_b128 %0, %1, off"
               :
               : "v"(lds_off), "v"((unsigned long long)(uintptr_t)gaddr)
               : "memory");
}

__device__ __forceinline__ void wait_async0() {
  asm volatile("s_wait_asynccnt 0x0" ::: "memory");
}

// one KSTEP panel of WMMAs: two 16-col tiles per wave
__device__ __forceinline__ void compute_panel(const float* __restrict__ aRow,
                                              const float* __restrict__ wPanel,
                                              int cBase, int lm, int hi,
                                              v8f& acc0, v8f& acc1) {
#pragma unroll
  for (int kk = 0; kk < KSTEP; kk += 4) {
    const int ka = kk + 2 * hi;
    v2f a;  a.x  = aRow[ka];                         a.y  = aRow[ka + 1];
    v2f b0; b0.x = wPanel[ka * MD + cBase + lm];      b0.y = wPanel[(ka + 1) * MD + cBase + lm];
    v2f b1; b1.x = wPanel[ka * MD + cBase + 16 + lm]; b1.y = wPanel[(ka + 1) * MD + cBase + 16 + lm];
    acc0 = wmma4(a, b0, acc0);
    acc1 = wmma4(a, b1, acc1);
  }
}

// ---------------------------------------------------------------------------
// zero accumulators (runs every launch -> deterministic)
// ---------------------------------------------------------------------------
__global__ void zero_kernel(float* a, size_t na, float* b, size_t nb,
                            float* c, size_t nc) {
  const size_t stride = (size_t)gridDim.x * blockDim.x;
  const size_t t0 = (size_t)blockIdx.x * blockDim.x + threadIdx.x;
  for (size_t i = t0; i < na; i += stride) a[i] = 0.f;
  for (size_t i = t0; i < nb; i += stride) b[i] = 0.f;
  for (size_t i = t0; i < nc; i += stride) c[i] = 0.f;
}

// ---------------------------------------------------------------------------
// Edge MLP: ek_mlp = phi_e([a1,a2,nbr]); fused residual + both segment sums.
// 256 threads (8 waves), 32 edges/block. Double-buffered async LDS staging.
// ---------------------------------------------------------------------------
__global__ __launch_bounds__(256) void edge_mlp_kernel(
    const float* __restrict__ atom, const float* __restrict__ nbr,
    const float* __restrict__ numn,
    const int* __restrict__ idx1, const int* __restrict__ idx2,
    const float* __restrict__ We1, const float* __restrict__ be1,
    const float* __restrict__ We2, const float* __restrict__ be2,
    const float* __restrict__ We3, const float* __restrict__ be3,
    float* __restrict__ out_ek, float* __restrict__ rho,
    float* __restrict__ eksum_anf,  // atom_nbr_fea base; [node*256+128+col]
    int E) {
  __shared__ float ldsA[2][EPB * LDA];
  __shared__ float ldsW[2][KSTEP * MD];
  __shared__ float ldsH[EPB * LDH];

  const int tid = threadIdx.x;
  const int lane = tid & 31;
  const int lm = lane & 15;
  const int hi = lane >> 4;
  const int wid = tid >> 5;
  const int m0 = (wid >> 2) << 4;
  const int cBase = (wid & 3) * 32;
  const int eBase = blockIdx.x * EPB;
  const int sE = tid >> 3;
  const int sK = (tid & 7) << 2;

  // clamped staging row (OOB rows produce garbage that the epilogue discards)
  int egs = eBase + sE; if (egs >= E) egs = E - 1;
  const int i1s = idx1[egs], i2s = idx2[egs];

  auto stageA = [&](int k0, float* buf) {
    const int kg = k0 + sK;
    const float* src;
    if (kg < 128)      src = atom + (size_t)i1s * MD + kg;
    else if (kg < 256) src = atom + (size_t)i2s * MD + (kg - 128);
    else               src = nbr + (size_t)egs * MD + (kg - 256);
    async_b128(lds_off32(buf + sE * LDA + sK), src);
  };
  auto stageW = [&](const float* W, int k0, float* buf) {
    const float* g = W + (size_t)k0 * MD + tid * 16;
    float* d = buf + tid * 16;
#pragma unroll
    for (int i = 0; i < 4; ++i) async_b128(lds_off32(d + i * 4), g + i * 4);
  };
  auto hstore = [&](const v8f& a0, const v8f& a1, const float* bias, bool act) {
    const int c0 = cBase + lm;
    const float bA = bias[c0], bB = bias[c0 + 16];
#pragma unroll
    for (int j = 0; j < 8; ++j) {
      const int row = m0 + j + hi * 8;
      float xA = a0[j] + bA, xB = a1[j] + bB;
      if (act) { xA = lrelu(xA); xB = lrelu(xB); }
      ldsH[row * LDH + c0]      = xA;
      ldsH[row * LDH + c0 + 16] = xB;
    }
  };

  v8f acc0 = vzero(), acc1 = vzero();

  // ---------------- layer 1: [32 x 384] @ We1 ----------------
  int cur = 0;
  stageA(0, ldsA[0]);
  stageW(We1, 0, ldsW[0]);
  for (int p = 0; p < 12; ++p) {
    wait_async0();
    __syncthreads();
    if (p + 1 < 12) {
      stageA((p + 1) * KSTEP, ldsA[cur ^ 1]);
      stageW(We1, (p + 1) * KSTEP, ldsW[cur ^ 1]);
    }
    compute_panel(&ldsA[cur][(m0 + lm) * LDA], ldsW[cur], cBase, lm, hi, acc0, acc1);
    cur ^= 1;
  }
  // buf0 free (last read at p=10, confirmed by p=11 top barrier)
  stageW(We2, 0, ldsW[0]);
  hstore(acc0, acc1, be1, true);   // visibility via next loop-top barrier

  // ---------------- layer 2: [32 x 128] @ We2 ----------------
  acc0 = vzero(); acc1 = vzero();
  cur = 0;
  for (int p = 0; p < 4; ++p) {
    wait_async0();
    __syncthreads();
    if (p + 1 < 4) stageW(We2, (p + 1) * KSTEP, ldsW[cur ^ 1]);
    compute_panel(&ldsH[(m0 + lm) * LDH + p * KSTEP], ldsW[cur], cBase, lm, hi, acc0, acc1);
    cur ^= 1;
  }
  __syncthreads();                 // all ldsH reads done before overwrite
  stageW(We3, 0, ldsW[0]);
  hstore(acc0, acc1, be2, true);

  // ---------------- layer 3: [32 x 128] @ We3 ----------------
  acc0 = vzero(); acc1 = vzero();
  cur = 0;
  for (int p = 0; p < 4; ++p) {
    wait_async0();
    __syncthreads();
    if (p + 1 < 4) stageW(We3, (p + 1) * KSTEP, ldsW[cur ^ 1]);
    compute_panel(&ldsH[(m0 + lm) * LDH + p * KSTEP], ldsW[cur], cBase, lm, hi, acc0, acc1);
    cur ^= 1;
  }

  // epilogue: ek residual + segment sums (rho += mlp/nn ; ek_sum += ek/nn)
  {
    const int c0 = cBase + lm;
    const float bA = be3[c0], bB = be3[c0 + 16];
#pragma unroll
    for (int j = 0; j < 8; ++j) {
      const int row = m0 + j + hi * 8;
      const int eg = eBase + row;
      if (eg < E) {
        const int i1 = idx1[eg];
        const float inv = 1.0f / numn[i1];
        const size_t eb = (size_t)eg * MD;
        const size_t nb1 = (size_t)i1 * MD;
        const size_t ab = (size_t)i1 * 256 + 128;

        float mlpA = acc0[j] + bA;
        float oA = nbr[eb + c0] + mlpA;
        out_ek[eb + c0] = oA;
        atomicAdd(&rho[nb1 + c0], mlpA * inv);
        atomicAdd(&eksum_anf[ab + c0], oA * inv);

        float mlpB = acc1[j] + bB;
        float oB = nbr[eb + c0 + 16] + mlpB;
        out_ek[eb + c0 + 16] = oB;
        atomicAdd(&rho[nb1 + c0 + 16], mlpB * inv);
        atomicAdd(&eksum_anf[ab + c0 + 16], oB * inv);
      }
    }
  }
}

// ---------------------------------------------------------------------------
// Node MLP: vi_raw = phi_v([atom, rho]); raw vi + BN partial sums
// ---------------------------------------------------------------------------
__global__ __launch_bounds__(256) void node_mlp_kernel(
    const float* __restrict__ atom, const float* __restrict__ rho,
    const float* __restrict__ Wv1, const float* __restrict__ bv1,
    const float* __restrict__ Wv2, const float* __restrict__ bv2,
    const float* __restrict__ Wv3, const float* __restrict__ bv3,
    float* __restrict__ viraw, float* __restrict__ bnsum,
    float* __restrict__ bnsq, int N) {
  __shared__ float ldsA[2][EPB * LDA];
  __shared__ float ldsW[2][KSTEP * MD];
  __shared__ float ldsH[EPB * LDH];

  const int tid = threadIdx.x;
  const int lane = tid & 31;
  const int lm = lane & 15;
  const int hi = lane >> 4;
  const int wid = tid >> 5;
  const int m0 = (wid >> 2) << 4;
  const int cBase = (wid & 3) * 32;
  const int nBase = blockIdx.x * EPB;
  const int sE = tid >> 3;
  const int sK = (tid & 7) << 2;

  int ngs = nBase + sE; if (ngs >= N) ngs = N - 1;

  auto stageA = [&](int k0, float* buf) {
    const int kg = k0 + sK;
    const float* src = (kg < 128) ? atom + (size_t)ngs * MD + kg
                                  : rho + (size_t)ngs * MD + (kg - 128);
    async_b128(lds_off32(buf + sE * LDA + sK), src);
  };
  auto stageW = [&](const float* W, int k0, float* buf) {
    const float* g = W + (size_t)k0 * MD + tid * 16;
    float* d = buf + tid * 16;
#pragma unroll
    for (int i = 0; i < 4; ++i) async_b128(lds_off32(d + i * 4), g + i * 4);
  };
  auto hstore = [&](const v8f& a0, const v8f& a1, const float* bias) {
    const int c0 = cBase + lm;
    const float bA = bias[c0], bB = bias[c0 + 16];
#pragma unroll
    for (int j = 0; j < 8; ++j) {
      const int row = m0 + j + hi * 8;
      ldsH[row * LDH + c0]      = lrelu(a0[j] + bA);
      ldsH[row * LDH + c0 + 16] = lrelu(a1[j] + bB);
    }
  };

  v8f acc0 = vzero(), acc1 = vzero();

  // ---------------- layer 1: [32 x 256] @ Wv1 ----------------
  int cur = 0;
  stageA(0, ldsA[0]);
  stageW(Wv1, 0, ldsW[0]);
  for (int p = 0; p < 8; ++p) {
    wait_async0();
    __syncthreads();
    if (p + 1 < 8) {
      stageA((p + 1) * KSTEP, ldsA[cur ^ 1]);
      stageW(Wv1, (p + 1) * KSTEP, ldsW[cur ^ 1]);
    }
    compute_panel(&ldsA[cur][(m0 + lm) * LDA], ldsW[cur], cBase, lm, hi, acc0, acc1);
    cur ^= 1;
  }
  stageW(Wv2, 0, ldsW[0]);
  hstore(acc0, acc1, bv1);

  // ---------------- layer 2 ----------------
  acc0 = vzero(); acc1 = vzero();
  cur = 0;
  for (int p = 0; p < 4; ++p) {
    wait_async0();
    __syncthreads();
    if (p + 1 < 4) stageW(Wv2, (p + 1) * KSTEP, ldsW[cur ^ 1]);
    compute_panel(&ldsH[(m0 + lm) * LDH + p * KSTEP], ldsW[cur], cBase, lm, hi, acc0, acc1);
    cur ^= 1;
  }
  __syncthreads();
  stageW(Wv3, 0, ldsW[0]);
  hstore(acc0, acc1, bv2);

  // ---------------- layer 3 ----------------
  acc0 = vzero(); acc1 = vzero();
  cur = 0;
  for (int p = 0; p < 4; ++p) {
    wait_async0();
    __syncthreads();
    if (p + 1 < 4) stageW(Wv3, (p + 1) * KSTEP, ldsW[cur ^ 1]);
    compute_panel(&ldsH[(m0 + lm) * LDH + p * KSTEP], ldsW[cur], cBase, lm, hi, acc0, acc1);
    cur ^= 1;
  }

  // epilogue: raw vi + BN partial sums
  {
    const int c0 = cBase + lm;
    const float bA = bv3[c0], bB = bv3[c0 + 16];
    float sA = 0.f, qA = 0.f, sB = 0.f, qB = 0.f;
#pragma unroll
    for (int j = 0; j < 8; ++j) {
      const int row = m0 + j + hi * 8;
      const int ng = nBase + row;
      if (ng < N) {
        const float xA = acc0[j] + bA;
        viraw[(size_t)ng * MD + c0] = xA;
        sA += xA; qA += xA * xA;
        const float xB = acc1[j] + bB;
        viraw[(size_t)ng * MD + c0 + 16] = xB;
        sB += xB; qB += xB * xB;
      }
    }
    atomicAdd(&bnsum[c0], sA);      atomicAdd(&bnsq[c0], qA);
    atomicAdd(&bnsum[c0 + 16], sB); atomicAdd(&bnsq[c0 + 16], qB);
  }
}

// ---------------------------------------------------------------------------
// BatchNorm finalize + residual + atom_nbr_fea concat + crystal aggregation
// ---------------------------------------------------------------------------
__global__ __launch_bounds__(256) void finalize_kernel(
    const float* __restrict__ atom, const float* __restrict__ bnsum,
    const float* __restrict__ bnsq, const float* __restrict__ gamma,
    const float* __restrict__ beta, const int* __restrict__ cidx,
    float* __restrict__ vi_io, float* __restrict__ anf, float* __restrict__ gf,
    int N) {
  const size_t idx = (size_t)blockIdx.x * blockDim.x + threadIdx.x;
  if (idx >= (size_t)N * MD) return;
  const int node = (int)(idx >> 7);
  const int col = (int)(idx & 127);
  const float invN = 1.0f / (float)N;
  const float mu = bnsum[col] * invN;
  const float var = bnsq[col] * invN - mu * mu;
  const float sc = gamma[col] * rsqrtf(var + 1e-5f);
  const float v = (vi_io[idx] - mu) * sc + beta[col] + atom[idx];
  vi_io[idx] = v;
  const int cry = cidx[node];
  anf[(size_t)node * 256 + col] = v;
  atomicAdd(&gf[(size_t)cry * 256 + col], v);
  const float es = anf[(size_t)node * 256 + 128 + col];  // accumulated ek_sum
  atomicAdd(&gf[(size_t)cry * 256 + 128 + col], es);
}

// ---------------------------------------------------------------------------
extern "C" void kernel_launch(void* const* d_in, const int* in_sizes, int n_in,
                              void* d_out, int out_size, void* d_ws,
                              size_t ws_size, hipStream_t stream) {
  (void)n_in; (void)out_size; (void)ws_size;
  const float* atom = (const float*)d_in[0];
  const float* nbr  = (const float*)d_in[1];
  const float* numn = (const float*)d_in[2];
  const int* idx1   = (const int*)d_in[3];
  const int* idx2   = (const int*)d_in[4];
  const int* cidx   = (const int*)d_in[5];
  const float* We1 = (const float*)d_in[6];  const float* be1 = (const float*)d_in[7];
  const float* We2 = (const float*)d_in[8];  const float* be2 = (const float*)d_in[9];
  const float* We3 = (const float*)d_in[10]; const float* be3 = (const float*)d_in[11];
  const float* Wv1 = (const float*)d_in[12]; const float* bv1 = (const float*)d_in[13];
  const float* Wv2 = (const float*)d_in[14]; const float* bv2 = (const float*)d_in[15];
  const float* Wv3 = (const float*)d_in[16]; const float* bv3 = (const float*)d_in[17];
  const float* gam = (const float*)d_in[18]; const float* bet = (const float*)d_in[19];

  const int N = in_sizes[0] / MD;   // 50000
  const int E = in_sizes[1] / MD;   // 600000
  const int NC = 256;

  float* out_ek  = (float*)d_out;                       // [E,128]
  float* out_vi  = out_ek + (size_t)E * MD;             // [N,128]
  float* out_gf  = out_vi + (size_t)N * MD;             // [256,256]
  float* out_anf = out_gf + (size_t)NC * 256;           // [N,256]

  float* rho   = (float*)d_ws;                          // [N,128]
  float* bnsum = rho + (size_t)N * MD;                  // [128]
  float* bnsq  = bnsum + MD;                            // [128]

  zero_kernel<<<2048, 256, 0, stream>>>(rho, (size_t)N * MD + 2 * MD,
                                        out_anf, (size_t)N * 256,
                                        out_gf, (size_t)NC * 256);

  edge_mlp_kernel<<<(E + EPB - 1) / EPB, 256, 0, stream>>>(
      atom, nbr, numn, idx1, idx2, We1, be1, We2, be2, We3, be3,
      out_ek, rho, out_anf, E);

  node_mlp_kernel<<<(N + EPB - 1) / EPB, 256, 0, stream>>>(
      atom, rho, Wv1, bv1, Wv2, bv2, Wv3, bv3, out_vi, bnsum, bnsq, N);

  finalize_kernel<<<(int)(((size_t)N * MD + 255) / 256), 256, 0, stream>>>(
      atom, bnsum, bnsq, gam, bet, cidx, out_vi, out_anf, out_gf, N);
}